// DicepolyTopk_48034914238680
// MI455X (gfx1250) — compile-verified
//
#include <hip/hip_runtime.h>
#include <math.h>

// ---------------------------------------------------------------------------
// DicePolyTopk for MI455X (gfx1250, wave32).
//
//  pass1 : stream p,t -> dice partial sums (WMMA f32 16x16x4 wave reduction)
//          + poly1 keys (float bits, monotonic since poly1 >= 0)
//          + LDS histogram of key bits [30:20]  (2048 bins)
//          + optional key cache in workspace
//  select: radix-select rounds walk histograms top-down to pin the exact
//          k-th largest key T (bits [30:20], [19:9], [8:0])
//  pass4 : sum of values with key > T (WMMA wave reduction), ties handled
//          exactly as (k_rem * T) since equal values are bit-identical.
//  finalize: dice + topk mean -> out[0]
// ---------------------------------------------------------------------------

typedef __attribute__((ext_vector_type(2))) float v2f;
typedef __attribute__((ext_vector_type(8))) float v8f;

#define POLY_EPS 3.1f
#define HBINS0 2048
#define HBINS1 2048
#define HBINS2 512
#define STATE_WORDS 16
#define TPB 256
#define NBLOCKS 1024

// noinline: guarantees bit-identical key computation across all passes in the
// recompute (non-cached) fallback path.
__device__ __attribute__((noinline)) float poly1_val(float p, float t) {
  float bce = -(t * logf(p) + (1.0f - t) * log1pf(-p));
  float pt  = expf(-bce);
  return bce + (1.0f - pt) * POLY_EPS;
}

// Wave-level f32 sum using V_WMMA_F32_16X16X4_F32.
// A: lane L's VGPR0 holds element (M = L%16, K = L<16 ? 0 : 2); VGPR1 = 0.
// B = all ones  =>  D[m][n] = partial_m + partial_{m+16}.
// D layout: lanes 0-15 hold M=0..7 in d[0..7]; lanes 16-31 hold M=8..15.
// Sum d[0..7] per lane, then fold lane n with lane n+16.
__device__ __forceinline__ float wave_reduce_wmma(float v) {
  v2f a; a[0] = v;    a[1] = 0.0f;
  v2f b; b[0] = 1.0f; b[1] = 1.0f;
  v8f c = {};
  v8f d = __builtin_amdgcn_wmma_f32_16x16x4_f32(
      /*neg_a=*/false, a, /*neg_b=*/false, b,
      /*c_mod=*/(short)0, c, /*reuse_a=*/false, /*reuse_b=*/false);
  float s = ((d[0] + d[1]) + (d[2] + d[3])) + ((d[4] + d[5]) + (d[6] + d[7]));
  s += __shfl_xor(s, 16, 32);
  return s;  // every lane holds the full wave sum
}

__global__ void zero_state_kernel(unsigned* __restrict__ u, int n) {
  int i = blockIdx.x * blockDim.x + threadIdx.x;
  if (i < n) u[i] = 0u;
}

// pass1: dice sums + key histogram (bits 30:20) + optional key cache
template <bool CACHED>
__global__ void pass1_kernel(const float* __restrict__ preds,
                             const float* __restrict__ gts,
                             float* __restrict__ sums,        // [0]=it [1]=i [2]=t
                             unsigned* __restrict__ hist0,
                             unsigned* __restrict__ keys,
                             int n4) {
  __shared__ unsigned h[HBINS0];
  for (int i = threadIdx.x; i < HBINS0; i += blockDim.x) h[i] = 0u;
  __syncthreads();

  float s_it = 0.0f, s_i = 0.0f, s_t = 0.0f;
  int tid    = blockIdx.x * blockDim.x + threadIdx.x;
  int stride = gridDim.x * blockDim.x;
  const float4* p4 = (const float4*)preds;
  const float4* t4 = (const float4*)gts;

  for (int i = tid; i < n4; i += stride) {
    float4 p = p4[i];
    float4 t = t4[i];
    s_it += p.x * t.x + p.y * t.y + p.z * t.z + p.w * t.w;
    s_i  += (p.x + p.y) + (p.z + p.w);
    s_t  += (t.x + t.y) + (t.z + t.w);
    unsigned k0 = __float_as_uint(poly1_val(p.x, t.x));
    unsigned k1 = __float_as_uint(poly1_val(p.y, t.y));
    unsigned k2 = __float_as_uint(poly1_val(p.z, t.z));
    unsigned k3 = __float_as_uint(poly1_val(p.w, t.w));
    if (CACHED) {
      uint4 kk; kk.x = k0; kk.y = k1; kk.z = k2; kk.w = k3;
      ((uint4*)keys)[i] = kk;
    }
    atomicAdd(&h[k0 >> 20], 1u);
    atomicAdd(&h[k1 >> 20], 1u);
    atomicAdd(&h[k2 >> 20], 1u);
    atomicAdd(&h[k3 >> 20], 1u);
  }

  // cross-lane reduce via WMMA, one global atomic per wave per quantity
  float w_it = wave_reduce_wmma(s_it);
  float w_i  = wave_reduce_wmma(s_i);
  float w_t  = wave_reduce_wmma(s_t);
  if ((threadIdx.x & 31) == 0) {
    atomicAdd(&sums[0], w_it);
    atomicAdd(&sums[1], w_i);
    atomicAdd(&sums[2], w_t);
  }

  __syncthreads();
  for (int i = threadIdx.x; i < HBINS0; i += blockDim.x) {
    unsigned c = h[i];
    if (c) atomicAdd(&hist0[i], c);
  }
}

// rounds 2/3: histogram of lower key bits for keys matching current prefix
template <bool CACHED>
__global__ void hist_pass_kernel(const float* __restrict__ preds,
                                 const float* __restrict__ gts,
                                 const unsigned* __restrict__ keys,
                                 const unsigned* __restrict__ state,
                                 unsigned* __restrict__ hist,
                                 int matchShift, int binShift, unsigned binMask,
                                 int n4) {
  __shared__ unsigned h[HBINS0];
  for (int i = threadIdx.x; i <= (int)binMask; i += blockDim.x) h[i] = 0u;
  __syncthreads();

  const unsigned want = state[5] >> matchShift;
  int tid    = blockIdx.x * blockDim.x + threadIdx.x;
  int stride = gridDim.x * blockDim.x;

  if (CACHED) {
    const uint4* k4 = (const uint4*)keys;
    for (int i = tid; i < n4; i += stride) {
      uint4 kk = k4[i];
      if ((kk.x >> matchShift) == want) atomicAdd(&h[(kk.x >> binShift) & binMask], 1u);
      if ((kk.y >> matchShift) == want) atomicAdd(&h[(kk.y >> binShift) & binMask], 1u);
      if ((kk.z >> matchShift) == want) atomicAdd(&h[(kk.z >> binShift) & binMask], 1u);
      if ((kk.w >> matchShift) == want) atomicAdd(&h[(kk.w >> binShift) & binMask], 1u);
    }
  } else {
    const float4* p4 = (const float4*)preds;
    const float4* t4 = (const float4*)gts;
    for (int i = tid; i < n4; i += stride) {
      float4 p = p4[i];
      float4 t = t4[i];
      unsigned kx = __float_as_uint(poly1_val(p.x, t.x));
      unsigned ky = __float_as_uint(poly1_val(p.y, t.y));
      unsigned kz = __float_as_uint(poly1_val(p.z, t.z));
      unsigned kw = __float_as_uint(poly1_val(p.w, t.w));
      if ((kx >> matchShift) == want) atomicAdd(&h[(kx >> binShift) & binMask], 1u);
      if ((ky >> matchShift) == want) atomicAdd(&h[(ky >> binShift) & binMask], 1u);
      if ((kz >> matchShift) == want) atomicAdd(&h[(kz >> binShift) & binMask], 1u);
      if ((kw >> matchShift) == want) atomicAdd(&h[(kw >> binShift) & binMask], 1u);
    }
  }

  __syncthreads();
  for (int i = threadIdx.x; i <= (int)binMask; i += blockDim.x) {
    unsigned c = h[i];
    if (c) atomicAdd(&hist[i], c);
  }
}

// top-down suffix scan of a histogram: pick the bin containing the k-th
// largest; chain k_rem / prefix through state[]. Tiny serial kernel.
__global__ void select_kernel(const unsigned* __restrict__ hist, int nbins,
                              int shift, int first,
                              unsigned* __restrict__ state, unsigned k0) {
  if (blockIdx.x == 0 && threadIdx.x == 0) {
    unsigned kneed = first ? k0 : state[4];
    unsigned cum = 0;
    int sel = 0;
    for (int b = nbins - 1; b >= 0; --b) {
      unsigned c = hist[b];
      if (cum + c >= kneed) { sel = b; break; }
      cum += c;
    }
    state[4] = kneed - cum;               // still needed from inside bin `sel`
    unsigned pfx = first ? 0u : state[5];
    pfx |= ((unsigned)sel) << shift;
    state[5] = pfx;
    state[6] = pfx;                        // after final round: exact threshold T
  }
}

// pass4: sum of poly1 values strictly above threshold T
template <bool CACHED>
__global__ void topk_sum_kernel(const float* __restrict__ preds,
                                const float* __restrict__ gts,
                                const unsigned* __restrict__ keys,
                                const unsigned* __restrict__ state,
                                float* __restrict__ sums,      // [3] = topk_gt_sum
                                int n4) {
  const unsigned T = state[6];
  float s = 0.0f;
  int tid    = blockIdx.x * blockDim.x + threadIdx.x;
  int stride = gridDim.x * blockDim.x;

  if (CACHED) {
    const uint4* k4 = (const uint4*)keys;
    for (int i = tid; i < n4; i += stride) {
      uint4 kk = k4[i];
      if (kk.x > T) s += __uint_as_float(kk.x);
      if (kk.y > T) s += __uint_as_float(kk.y);
      if (kk.z > T) s += __uint_as_float(kk.z);
      if (kk.w > T) s += __uint_as_float(kk.w);
    }
  } else {
    const float4* p4 = (const float4*)preds;
    const float4* t4 = (const float4*)gts;
    for (int i = tid; i < n4; i += stride) {
      float4 p = p4[i];
      float4 t = t4[i];
      unsigned kx = __float_as_uint(poly1_val(p.x, t.x));
      unsigned ky = __float_as_uint(poly1_val(p.y, t.y));
      unsigned kz = __float_as_uint(poly1_val(p.z, t.z));
      unsigned kw = __float_as_uint(poly1_val(p.w, t.w));
      if (kx > T) s += __uint_as_float(kx);
      if (ky > T) s += __uint_as_float(ky);
      if (kz > T) s += __uint_as_float(kz);
      if (kw > T) s += __uint_as_float(kw);
    }
  }

  float w = wave_reduce_wmma(s);
  if ((threadIdx.x & 31) == 0) atomicAdd(&sums[3], w);
}

__global__ void finalize_kernel(const float* __restrict__ sums,
                                const unsigned* __restrict__ state,
                                float* __restrict__ out, unsigned kcount) {
  if (blockIdx.x == 0 && threadIdx.x == 0) {
    float dice = 1.0f - (2.0f * sums[0] + 1.0f) / (sums[1] + sums[2] + 1.0f);
    float tv   = __uint_as_float(state[6]);   // exact k-th largest value
    float ties = (float)state[4];             // #elements equal to T in top-k
    float mean = (sums[3] + ties * tv) / (float)kcount;
    out[0] = dice + mean;
  }
}

extern "C" void kernel_launch(void* const* d_in, const int* in_sizes, int n_in,
                              void* d_out, int out_size, void* d_ws, size_t ws_size,
                              hipStream_t stream) {
  const float* preds = (const float*)d_in[0];
  const float* gts   = (const float*)d_in[1];
  float* out         = (float*)d_out;

  const int N  = in_sizes[0];
  const int n4 = N / 4;                                   // N = 64*512*512, /4 exact
  const unsigned kcount = (unsigned)((long long)N * 10 / 100);

  unsigned* u     = (unsigned*)d_ws;
  float*    sums  = (float*)d_ws;                         // u[0..3] as float
  unsigned* hist0 = u + STATE_WORDS;
  unsigned* hist1 = hist0 + HBINS0;
  unsigned* hist2 = hist1 + HBINS1;
  unsigned* keys  = (unsigned*)((char*)d_ws + 65536);     // 16B-aligned key cache
  const bool cached = ws_size >= (size_t)65536 + (size_t)N * 4u;

  const int nzero = STATE_WORDS + HBINS0 + HBINS1 + HBINS2;
  zero_state_kernel<<<(nzero + TPB - 1) / TPB, TPB, 0, stream>>>(u, nzero);

  if (cached) {
    pass1_kernel<true><<<NBLOCKS, TPB, 0, stream>>>(preds, gts, sums, hist0, keys, n4);
    select_kernel<<<1, 1, 0, stream>>>(hist0, HBINS0, 20, 1, u, kcount);
    hist_pass_kernel<true><<<NBLOCKS, TPB, 0, stream>>>(preds, gts, keys, u, hist1,
                                                        20, 9, 0x7FFu, n4);
    select_kernel<<<1, 1, 0, stream>>>(hist1, HBINS1, 9, 0, u, kcount);
    hist_pass_kernel<true><<<NBLOCKS, TPB, 0, stream>>>(preds, gts, keys, u, hist2,
                                                        9, 0, 0x1FFu, n4);
    select_kernel<<<1, 1, 0, stream>>>(hist2, HBINS2, 0, 0, u, kcount);
    topk_sum_kernel<true><<<NBLOCKS, TPB, 0, stream>>>(preds, gts, keys, u, sums, n4);
  } else {
    pass1_kernel<false><<<NBLOCKS, TPB, 0, stream>>>(preds, gts, sums, hist0, nullptr, n4);
    select_kernel<<<1, 1, 0, stream>>>(hist0, HBINS0, 20, 1, u, kcount);
    hist_pass_kernel<false><<<NBLOCKS, TPB, 0, stream>>>(preds, gts, nullptr, u, hist1,
                                                         20, 9, 0x7FFu, n4);
    select_kernel<<<1, 1, 0, stream>>>(hist1, HBINS1, 9, 0, u, kcount);
    hist_pass_kernel<false><<<NBLOCKS, TPB, 0, stream>>>(preds, gts, nullptr, u, hist2,
                                                         9, 0, 0x1FFu, n4);
    select_kernel<<<1, 1, 0, stream>>>(hist2, HBINS2, 0, 0, u, kcount);
    topk_sum_kernel<false><<<NBLOCKS, TPB, 0, stream>>>(preds, gts, nullptr, u, sums, n4);
  }

  finalize_kernel<<<1, 1, 0, stream>>>(sums, u, out, kcount);
}